// IrrepsLinear_7232724927038
// MI455X (gfx1250) — compile-verified
//
#include <hip/hip_runtime.h>

typedef __attribute__((ext_vector_type(2))) float v2f;
typedef __attribute__((ext_vector_type(4))) float v4f;
typedef __attribute__((ext_vector_type(8))) float v8f;
typedef __attribute__((ext_vector_type(4))) int   v4i;

#define M_TOT   16
#define C_INOUT 128
#define LDS_ROW 132   // 528B row stride: 16B-aligned, rows shift 4 banks (64-bank LDS)
#define TILES   5     // node-tiles (of 16 nodes) per block; 50000 = 625 * 80

#if __has_builtin(__builtin_amdgcn_global_load_async_to_lds_b128)
#define USE_ASYNC 1
#else
#define USE_ASYNC 0
#endif

#define AS_GLOBAL __attribute__((address_space(1)))
#define AS_LDS    __attribute__((address_space(3)))

template <bool FULL>
__global__ __launch_bounds__(256)
void irreps_linear_wmma(const float* __restrict__ x,
                        const float* __restrict__ w,
                        const int*   __restrict__ seg,
                        float*       __restrict__ out,
                        int n_nodes, int super_base)
{
    __shared__ float Alds[TILES * 16 * LDS_ROW];

    const int blk   = blockIdx.x;
    const int m     = blk & 15;                  // spherical component
    const int super = (blk >> 4) + super_base;   // 80-node super-tile
    const int n0    = super * (16 * TILES);

    const int tid  = threadIdx.x;
    const int lane = tid & 31;
    const int wave = tid >> 5;                   // 0..7, owns cols [16w, 16w+16)
    const int path = seg[m];                     // uniform -> scalar load

    // ---- stage TILES A tiles: x[n0 .. n0+16*TILES-1, m, 0:128] -> LDS
    {
        const int row = tid >> 4;                // 0..15
        const int c0  = (tid & 15) * 8;          // 0..120 (32B-aligned)
        #pragma unroll
        for (int t = 0; t < TILES; ++t) {
            const int nn = n0 + t * 16 + row;
            const int rn = FULL ? nn : ((nn < n_nodes) ? nn : (n_nodes - 1));
            const float* src = x + (((size_t)rn * M_TOT + m) * C_INOUT + c0);
            float*       dst = &Alds[(t * 16 + row) * LDS_ROW + c0];
#if USE_ASYNC
            // async global->LDS, no VGPR round-trip; IOFFSET adds to both addrs
            __builtin_amdgcn_global_load_async_to_lds_b128(
                (AS_GLOBAL v4i*)src, (AS_LDS v4i*)dst, 0, 0);
            __builtin_amdgcn_global_load_async_to_lds_b128(
                (AS_GLOBAL v4i*)src, (AS_LDS v4i*)dst, 16, 0);
#else
            v4f a0 = *(const v4f*)(src);
            v4f a1 = *(const v4f*)(src + 4);
            *(v4f*)(dst)     = a0;
            *(v4f*)(dst + 4) = a1;
#endif
        }
    }

    const int col     = lane & 15;
    const int hi      = lane >> 4;               // 0: lanes 0-15, 1: lanes 16-31
    const int colbase = wave * 16;

    // ---- preload B slice W[path][0:128][colbase+col] into 64 VGPRs
    // B frag for k-step kk: v0 = W[4kk+2*hi][col], v1 = W[4kk+2*hi+1][col]
    const float* wp = w + ((size_t)path * C_INOUT * C_INOUT) + colbase + col;
    float b0[32], b1[32];
    #pragma unroll
    for (int kk = 0; kk < 32; ++kk) {
        const int k = 4 * kk + 2 * hi;
        b0[kk] = wp[(size_t)(k    ) * C_INOUT];
        b1[kk] = wp[(size_t)(k + 1) * C_INOUT];
    }

#if USE_ASYNC
  #if __has_builtin(__builtin_amdgcn_s_wait_asynccnt)
    __builtin_amdgcn_s_wait_asynccnt(0);
  #else
    asm volatile("s_wait_asynccnt 0" ::: "memory");
  #endif
#endif
    __syncthreads();

    // ---- per node-tile: 32 x v_wmma_f32_16x16x4_f32, B regs reused TILES x
    const int arow = (lane & 15) * LDS_ROW + 2 * hi;
    float* op = out + (((size_t)(n0 + 8 * hi) * M_TOT + m) * C_INOUT + colbase + col);

    for (int t = 0; t < TILES; ++t) {
        v8f acc = {};
        const float* ap = &Alds[t * 16 * LDS_ROW + arow];
        #pragma unroll
        for (int kk = 0; kk < 32; ++kk) {
            v2f a = *(const v2f*)(ap + 4 * kk);  // ds_load_b64 (2addr-paired)
            v2f b = { b0[kk], b1[kk] };
            acc = __builtin_amdgcn_wmma_f32_16x16x4_f32(
                      /*neg_a=*/false, a, /*neg_b=*/false, b,
                      /*c_mod=*/(short)0, acc, /*reuse_a=*/false, /*reuse_b=*/false);
        }

        // store: acc[j] -> out[n0 + 16t + j + 8*hi][m][colbase + col]
        #pragma unroll
        for (int j = 0; j < 8; ++j) {
            if (FULL) {
                op[(size_t)(t * 16 + j) * (M_TOT * C_INOUT)] = acc[j];
            } else {
                const int rn = n0 + t * 16 + j + 8 * hi;
                if (rn < n_nodes) {
                    out[((size_t)rn * M_TOT + m) * C_INOUT + colbase + col] = acc[j];
                }
            }
        }
    }
}

extern "C" void kernel_launch(void* const* d_in, const int* in_sizes, int n_in,
                              void* d_out, int out_size, void* d_ws, size_t ws_size,
                              hipStream_t stream) {
    const float* x   = (const float*)d_in[0];   // (N, 16, 128) f32
    const float* w   = (const float*)d_in[1];   // (4, 128, 128) f32
    const int*   seg = (const int*)d_in[2];     // (16,) i32
    float*       out = (float*)d_out;           // (N, 16, 128) f32

    const int n_nodes = in_sizes[0] / (M_TOT * C_INOUT);
    const int n_full  = n_nodes / (16 * TILES);         // full 80-node super-tiles
    const int rem     = n_nodes - n_full * (16 * TILES);

    if (n_full > 0) {
        irreps_linear_wmma<true><<<dim3(n_full * M_TOT), dim3(256), 0, stream>>>(
            x, w, seg, out, n_nodes, 0);
    }
    if (rem > 0) {
        irreps_linear_wmma<false><<<dim3(M_TOT), dim3(256), 0, stream>>>(
            x, w, seg, out, n_nodes, n_full);
    }
}